// InPlaneLocallyConnected2D_14370960573268
// MI455X (gfx1250) — compile-verified
//
#include <hip/hip_runtime.h>
#include <hip/hip_bf16.h>
#include <stdint.h>

// Problem constants (from reference): x:(32,96,96,128) f32, kernel:(94*94,9,1) f32
#define KH   3
#define KW   3
#define B_   32
#define H_   96
#define W_   96
#define C_   128
#define OR_  (H_ - KH + 1)   // 94
#define OC_  (W_ - KW + 1)   // 94
#define JT   8               // output columns per block (one per wave)
#define NJT  ((OC_ + JT - 1) / JT)   // 12
#define TCOLS (JT + KW - 1)  // 10 input columns staged
#define C4   (C_ / 4)        // 32 float4 per pixel == one wave32

#define NTHREADS 256

typedef float v4f __attribute__((ext_vector_type(4)));
typedef int   v4i __attribute__((ext_vector_type(4)));

#if defined(__gfx1250__)
typedef v4i __attribute__((address_space(1)))* gptr_v4i;   // global
typedef v4i __attribute__((address_space(3)))* lptr_v4i;   // LDS
#endif

__global__ __launch_bounds__(NTHREADS)
void ilc2d_kernel(const float* __restrict__ x,
                  const float* __restrict__ kern,
                  float* __restrict__ out)
{
    // x tile: rows i..i+2, cols j0..j0+9, all 128 channels  (15,360 B)
    __shared__ __align__(16) float xs[KH * TCOLS * C_];
    // per-block weights: 8 output cols x 9 taps
    __shared__ float ws[JT * KH * KW];

    const int tid = threadIdx.x;
    int blk = blockIdx.x;
    const int jt = blk % NJT;  blk /= NJT;
    const int i  = blk % OR_;  blk /= OR_;
    const int b  = blk;
    const int j0 = jt * JT;

    // ---- stage x window into LDS via CDNA5 async global->LDS copies ----
    const int totalV4 = KH * TCOLS * C4;   // 960 x 16B transfers
    for (int k = tid; k < totalV4; k += NTHREADS) {
        const int c4  = k & (C4 - 1);
        const int col = (k / C4) % TCOLS;
        const int row = k / (C4 * TCOLS);
        int gcol = j0 + col;
        if (gcol > W_ - 1) gcol = W_ - 1;          // halo clamp (clamped cols never used)
        const float* gsrc = x + (((size_t)(b * H_ + (i + row)) * W_ + gcol) * C_ + c4 * 4);
        float*       ldst = &xs[(row * TCOLS + col) * C_ + c4 * 4];
#if defined(__gfx1250__) && __has_builtin(__builtin_amdgcn_global_load_async_to_lds_b128)
        __builtin_amdgcn_global_load_async_to_lds_b128(
            (gptr_v4i)const_cast<float*>(gsrc),
            (lptr_v4i)ldst,
            /*imm offset*/0, /*cpol*/0);
#else
        *(v4f*)ldst = *(const v4f*)gsrc;           // fallback: synchronous copy
#endif
    }

    // ---- stage the 8x9 weights (normal LDS stores; tiny) ----
    if (tid < JT * KH * KW) {
        const int jw = tid / (KH * KW);
        const int tw = tid % (KH * KW);
        const int j  = j0 + jw;
        ws[tid] = (j < OC_) ? kern[((size_t)i * OC_ + j) * (KH * KW) + tw] : 0.0f;
    }

#if defined(__gfx1250__) && __has_builtin(__builtin_amdgcn_global_load_async_to_lds_b128)
  #if __has_builtin(__builtin_amdgcn_s_wait_asynccnt)
    __builtin_amdgcn_s_wait_asynccnt(0);
  #else
    asm volatile("s_wait_asynccnt 0" ::: "memory");
  #endif
#endif
    __syncthreads();

    // ---- compute: one wave per output pixel, lane = 4 channels ----
    const int w    = tid >> 5;      // wave id -> j offset within tile
    const int lane = tid & 31;      // channel group
    const int j    = j0 + w;
    if (j < OC_) {
        v4f acc = {0.f, 0.f, 0.f, 0.f};
#pragma unroll
        for (int di = 0; di < KH; ++di) {
#pragma unroll
            for (int dj = 0; dj < KW; ++dj) {
                const float wv = ws[w * (KH * KW) + di * KW + dj];    // wave-uniform -> LDS broadcast
                const v4f   xv = *(const v4f*)&xs[(di * TCOLS + (w + dj)) * C_ + lane * 4];
                acc += xv * wv;
            }
        }
        float* op = out + (((size_t)(b * OR_ + i) * OC_ + j) * C_ + lane * 4);
        __builtin_nontemporal_store(acc, (v4f*)op);   // write-once output: TH=NT
    }
}

extern "C" void kernel_launch(void* const* d_in, const int* in_sizes, int n_in,
                              void* d_out, int out_size, void* d_ws, size_t ws_size,
                              hipStream_t stream) {
    (void)in_sizes; (void)n_in; (void)out_size; (void)d_ws; (void)ws_size;
    const float* x    = (const float*)d_in[0];
    const float* kern = (const float*)d_in[1];
    float*       out  = (float*)d_out;

    const int nblocks = B_ * OR_ * NJT;   // 32 * 94 * 12 = 36,096
    ilc2d_kernel<<<nblocks, NTHREADS, 0, stream>>>(x, kern, out);
}